// RVQE_90950227460859
// MI455X (gfx1250) — compile-verified
//
#include <hip/hip_runtime.h>
#include <hip/hip_bf16.h>

typedef __attribute__((ext_vector_type(2))) float v2f;
typedef __attribute__((ext_vector_type(8))) float v8f;

// Problem constants (fixed by the reference)
#define WS_Q   10
#define NQ     12
#define NSTATE 4096          // 2^12 amplitudes per batch element
#define T_STEPS 8            // T-1 scan steps
#define IN_Q   6

// -------- state-index mapping for the 3 WMMA "parts" of the RY layer --------
// part 0: qubits 0..3  -> state bits 11..8 : idx = j*256 + (tt*16+cl)
// part 1: qubits 4..7  -> state bits  7..4 : idx = tt*256 + j*16 + cl
// part 2: qubits 8..9 (x I4 on ancillas)   : idx = (tt*16+cl)*16 + j
__device__ __forceinline__ int sidx(int part, int tt, int j, int cl) {
  if (part == 0) return (j << 8) + (tt << 4) + cl;
  if (part == 1) return (tt << 8) + (j << 4) + cl;
  return (((tt << 4) + cl) << 4) + j;
}

// Element [j][k] of the 16x16 tensor-product RY matrix for one part.
// RY(theta): p0' = c*p0 - s*p1 ; p1' = s*p0 + c*p1  => R[a][b]: R00=c,R01=-s,R10=s,R11=c
__device__ __forceinline__ float mat_entry(const float* ua_s, int part, int j, int k) {
  float v = 1.f;
#pragma unroll
  for (int q = 0; q < 4; ++q) {
    int ja = (j >> (3 - q)) & 1;
    int kb = (k >> (3 - q)) & 1;
    int qubit = part * 4 + q;
    if (qubit < WS_Q) {
      float h = 0.5f * ua_s[qubit];
      float c = cosf(h), s = sinf(h);
      v *= (ja == kb) ? c : (ja ? s : -s);
    } else {
      if (ja != kb) v = 0.f;   // identity factor on ancilla bits (part 2)
    }
  }
  return v;
}

// Apply one 16x16 matrix to the state as M(16x16) x S(16x256) using chained
// f32 WMMA 16x16x4 (K accumulated over 4 steps). Each wave owns 2 column tiles.
__device__ __forceinline__ void apply_ry_wmma(float* psi, const float* M,
                                              int part, int lane, int wave) {
  const int cl = lane & 15;   // column within tile; also A-matrix row (M = lane%16)
  const int hi = lane >> 4;   // lane half selects K-subrows {0,1} vs {2,3}
#pragma unroll
  for (int ti = 0; ti < 2; ++ti) {
    const int tt = wave * 2 + ti;          // tile 0..15
    v8f acc = {};
#pragma unroll
    for (int t = 0; t < 4; ++t) {
      const int kb = 4 * t + 2 * hi;       // global K row pair for this lane half
      v2f a, b;
      a.x = M[cl * 16 + kb];               // A[row=lane%16][kb]
      a.y = M[cl * 16 + kb + 1];
      b.x = psi[sidx(part, tt, kb, cl)];   // B[kb][col=cl]
      b.y = psi[sidx(part, tt, kb + 1, cl)];
      acc = __builtin_amdgcn_wmma_f32_16x16x4_f32(
          /*neg_a=*/false, a, /*neg_b=*/false, b,
          /*c_mod=*/(short)0, acc, /*reuse_a=*/false, /*reuse_b=*/false);
    }
    // D layout: VGPR r holds row r (lanes 0-15) / row r+8 (lanes 16-31), col = cl
#pragma unroll
    for (int r = 0; r < 8; ++r)
      psi[sidx(part, tt, hi * 8 + r, cl)] = acc[r];
  }
}

// One "neuron" (order-2 construction) acting on disjoint 8-amplitude groups:
// group = two contiguous float4 blocks (ancillas are the 2 LSBs) paired across
// the outlane bit. U(s) == RY(s*phi_c) on q10 with phi_c = th[10] + sum c_i th[i].
__device__ __forceinline__ void neuron(float* psi, const float* th,
                                       int outlane, int tid) {
  const int ob = 9 - outlane;              // outlane's bit within the 10-bit block idx
  const int lowmask = (1 << ob) - 1;
#pragma unroll
  for (int gi = 0; gi < 2; ++gi) {
    const int g  = tid + gi * 256;                       // 512 groups total
    const int c0 = ((g & ~lowmask) << 1) | (g & lowmask); // insert 0 at bit ob
    const int c1 = c0 | (1 << ob);

    float phi0 = th[WS_Q];
#pragma unroll
    for (int i = 0; i < WS_Q; ++i)
      phi0 += ((c0 >> (9 - i)) & 1) ? th[i] : 0.f;
    const float phi1 = phi0 + th[outlane];

    float4 v0 = *reinterpret_cast<float4*>(&psi[c0 * 4]);
    float4 v1 = *reinterpret_cast<float4*>(&psi[c1 * 4]);
    float p[2][4] = {{v0.x, v0.y, v0.z, v0.w}, {v1.x, v1.y, v1.z, v1.w}};
    const float cs[2] = {cosf(0.5f * phi0), cosf(0.5f * phi1)};
    const float sn[2] = {sinf(0.5f * phi0), sinf(0.5f * phi1)};

    auto U = [&](float sign) {
#pragma unroll
      for (int o = 0; o < 2; ++o) {
        const float C = cs[o], S = sign * sn[o];
#pragma unroll
        for (int a1 = 0; a1 < 2; ++a1) {       // pair over a0 (q10), a = a0*2+a1
          float x = p[o][a1], y = p[o][2 + a1];
          p[o][a1]     = C * x - S * y;
          p[o][2 + a1] = S * x + C * y;
        }
      }
    };

    U(+1.f);
    // C(q10=1)-iY(q11, +1): (p_{a1=0}, p_{a1=1}) -> (p1, -p0)
#pragma unroll
    for (int o = 0; o < 2; ++o) { float x = p[o][2], y = p[o][3]; p[o][2] = y;  p[o][3] = -x; }
    U(-1.f);
    // C(q11=1)-iY(outlane, +1): (p_{o=0}, p_{o=1}) -> (p1, -p0)
#pragma unroll
    for (int a0 = 0; a0 < 2; ++a0) { int a = a0 * 2 + 1; float x = p[0][a], y = p[1][a]; p[0][a] = y; p[1][a] = -x; }
    U(+1.f);
    // C(q10=1)-iY(q11, -1): (p0, p1) -> (-p1, p0)
#pragma unroll
    for (int o = 0; o < 2; ++o) { float x = p[o][2], y = p[o][3]; p[o][2] = -y; p[o][3] = x; }
    U(-1.f);

    *reinterpret_cast<float4*>(&psi[c0 * 4]) = make_float4(p[0][0], p[0][1], p[0][2], p[0][3]);
    *reinterpret_cast<float4*>(&psi[c1 * 4]) = make_float4(p[1][0], p[1][1], p[1][2], p[1][3]);
  }
}

// X-gates on the 6 input qubits == permutation idx -> idx ^ (mask<<6)
__device__ __forceinline__ void xor_pass(float* psi, int xm, int tid) {
  float tmp[16];
#pragma unroll
  for (int k = 0; k < 16; ++k) tmp[k] = psi[tid * 16 + k];
  __syncthreads();
#pragma unroll
  for (int k = 0; k < 16; ++k) psi[(tid * 16 + k) ^ xm] = tmp[k];
  __syncthreads();
}

__global__ __launch_bounds__(256) void rvqe_kernel(const int* __restrict__ inputs,
                                                   const float* __restrict__ ua,
                                                   const float* __restrict__ nth,
                                                   float* __restrict__ out,
                                                   int out_size) {
  __shared__ __align__(16) float psi[NSTATE];     // 16 KB state, resident all steps
  __shared__ float Mlds[2][3][256];               // 6 precomputed 16x16 RY matrices
  __shared__ float th_lds[2][WS_Q][WS_Q + 1];     // neuron thetas
  __shared__ float probs_lds[64];
  __shared__ int   m9[9];                         // per-timestep 6-bit masks

  const int tid  = threadIdx.x;
  const int b    = blockIdx.x;
  const int lane = tid & 31;
  const int wave = tid >> 5;

  // |0...0> initial state
#pragma unroll
  for (int k = 0; k < 16; ++k) psi[tid * 16 + k] = 0.f;
  if (tid == 0) psi[0] = 1.f;

  // RY-layer matrices (same for all steps; depend only on stage angles)
  {
    const int j = tid >> 4, kk = tid & 15;
    for (int s = 0; s < 2; ++s)
      for (int part = 0; part < 3; ++part)
        Mlds[s][part][tid] = mat_entry(ua + s * WS_Q, part, j, kk);
  }
  // neuron thetas -> LDS
  {
    float* thf = &th_lds[0][0][0];
    for (int i = tid; i < 2 * WS_Q * (WS_Q + 1); i += 256) thf[i] = nth[i];
  }
  // input bit masks (qubit i -> bit 5-i of the top-6 block index)
  if (tid < 9) {
    int m = 0;
    for (int i = 0; i < IN_Q; ++i)
      m |= (inputs[(b * 9 + tid) * IN_Q + i] ? 1 : 0) << (5 - i);
    m9[tid] = m;
  }
  // measured = inputs[:,1:] (second tuple output), as float
  if (tid < T_STEPS * IN_Q && out_size >= 64 * T_STEPS * 64 + 64 * T_STEPS * IN_Q) {
    int t = tid / IN_Q, i = tid % IN_Q;
    out[64 * T_STEPS * 64 + (b * T_STEPS + t) * IN_Q + i] =
        (float)inputs[(b * 9 + t + 1) * IN_Q + i];
  }
  __syncthreads();

  for (int t = 0; t < T_STEPS; ++t) {
    // conditional flips from input bits
    xor_pass(psi, m9[t] << 6, tid);

    for (int s = 0; s < 2; ++s) {
      // RY layer: three 16x16 GEMM passes over the state via f32 WMMA chains
      for (int part = 0; part < 3; ++part) {
        apply_ry_wmma(psi, Mlds[s][part], part, lane, wave);
        __syncthreads();
      }
      // 10 neurons, each a single disjoint-group pass
      for (int ol = 0; ol < WS_Q; ++ol) {
        neuron(psi, th_lds[s][ol], ol, tid);
        __syncthreads();
      }
    }

    // measurement probabilities over the top-6 qubits
    if (tid < 64) {
      float acc = 0.f;
      const float* blk = &psi[tid * 64];
      for (int l = 0; l < 64; ++l) { float v = blk[l]; acc += v * v; }
      probs_lds[tid] = acc;
      out[(b * T_STEPS + t) * 64 + tid] = acc;
    }
    __syncthreads();

    // project onto target bits, renormalize (norm^2 == probs[target]), flip back
    const int tm = m9[t + 1];
    const float inv = rsqrtf(probs_lds[tm]);
    float tmp[16];
#pragma unroll
    for (int k = 0; k < 16; ++k) {
      int idx = tid * 16 + k;
      tmp[k] = ((idx >> 6) == tm) ? psi[idx] * inv : 0.f;
    }
    __syncthreads();
    const int xm = tm << 6;
#pragma unroll
    for (int k = 0; k < 16; ++k) psi[(tid * 16 + k) ^ xm] = tmp[k];
    __syncthreads();
  }
}

extern "C" void kernel_launch(void* const* d_in, const int* in_sizes, int n_in,
                              void* d_out, int out_size, void* d_ws, size_t ws_size,
                              hipStream_t stream) {
  const int*   inputs = (const int*)d_in[0];    // (B, 9, 6) int32
  const float* ua     = (const float*)d_in[1];  // (2, 10) f32
  const float* nth    = (const float*)d_in[2];  // (2, 10, 11) f32
  float* out = (float*)d_out;
  const int B = in_sizes[0] / (9 * IN_Q);       // 64
  rvqe_kernel<<<B, 256, 0, stream>>>(inputs, ua, nth, out, out_size);
}